// AddRelativePositionalEmbedding_31619549233181
// MI455X (gfx1250) — compile-verified
//
#include <hip/hip_runtime.h>
#include <hip/hip_bf16.h>

typedef __attribute__((ext_vector_type(2))) float v2f;
typedef __attribute__((ext_vector_type(8))) float v8f;

// ---------------------------------------------------------------------------
// Stage 1: rel = batched 64x64x64 GEMM using V_WMMA_F32_16X16X4_F32.
//
// For block p = b*64 + f:
//   A[m, c]  = q  [ b*262144 + f*f_stride + m*m_stride + c ]   (64x64, contig c)
//   B[c, n]  = rel_pos[ (f - n + 63)*64 + c ]                   (64x64)
//   O[m, n]  = out[ b*262144 + f*f_stride + m*m_stride + n ]
//
// rel_h launch: f = h, f_stride = 4096, m_stride = 64   (m iterates w)
// rel_w launch: f = w, f_stride = 64,   m_stride = 4096 (m iterates h)
// Both launches write the same (B, 64, 64, 64) row-major layout, i.e.
// rel[b,h,w,k] at linear offset ((b*64+h)*64+w)*64 + k.
//
// 256 threads = 8 waves. 16 output tiles of 16x16; each wave owns 2 tiles
// that share the A operand (same mtile, adjacent ntiles).
// WMMA F32 16x16x4 operand layout (ISA 7.12.2):
//   A (16x4):  lane L<16 -> row M=L holds {K0,K1}; L>=16 -> row M=L-16 {K2,K3}
//   B (4x16):  lane L<16 -> col N=L holds {K0,K1}; L>=16 -> col N=L-16 {K2,K3}
//   C/D (16x16): reg r, lane L -> M = r + (L>=16 ? 8 : 0), N = L&15
// ---------------------------------------------------------------------------
__global__ __launch_bounds__(256) void relpos_gemm_wmma(
    const float* __restrict__ q, const float* __restrict__ rel_pos,
    float* __restrict__ out, int f_stride, int m_stride)
{
  const int p = blockIdx.x;
  const int b = p >> 6;
  const int f = p & 63;
  const long base = (long)b * 262144 + (long)f * f_stride;

  const int wave = threadIdx.x >> 5;
  const int lane = threadIdx.x & 31;
  const int l15  = lane & 15;
  const int chalf = (lane >> 4) << 1;          // 0 for lanes 0-15, 2 for 16-31

  const int mtile = (wave >> 1) << 4;          // 0,16,32,48
  const int nbase = (wave & 1) << 5;           // 0 or 32
  const int n0 = nbase + l15;                  // B-col for tile 0
  const int n1 = n0 + 16;                      // B-col for tile 1

  const float* __restrict__ arow  = q + base + (long)(mtile + l15) * m_stride + chalf;
  const float* __restrict__ brow0 = rel_pos + (long)(f - n0 + 63) * 64 + chalf;
  const float* __restrict__ brow1 = rel_pos + (long)(f - n1 + 63) * 64 + chalf;

  v8f acc0 = {};
  v8f acc1 = {};

  #pragma unroll
  for (int c0 = 0; c0 < 64; c0 += 4) {
    v2f a;  a.x  = arow[c0];  a.y  = arow[c0 + 1];    // contiguous -> b64 load
    v2f b0; b0.x = brow0[c0]; b0.y = brow0[c0 + 1];
    v2f b1; b1.x = brow1[c0]; b1.y = brow1[c0 + 1];
    // (neg_a, A, neg_b, B, c_mod, C, reuse_a, reuse_b)
    acc0 = __builtin_amdgcn_wmma_f32_16x16x4_f32(false, a, false, b0,
                                                 (short)0, acc0, false, false);
    acc1 = __builtin_amdgcn_wmma_f32_16x16x4_f32(false, a, false, b1,
                                                 (short)0, acc1, false, false);
  }

  float* __restrict__ O = out + base;
  const int moff = mtile + ((lane >> 4) << 3);   // +8 for upper half-wave
  #pragma unroll
  for (int r = 0; r < 8; ++r) {
    const long mrow = (long)(moff + r) * m_stride;
    O[mrow + nbase + l15]      = acc0[r];
    O[mrow + nbase + 16 + l15] = acc1[r];
  }
}

// ---------------------------------------------------------------------------
// Stage 2: stream attention_map (512 MiB) once, add rel_h[kh] + rel_w[kw].
// One block per (b,h,w) slice of 64x64 floats = 1024 float4; 256 thr x 4 iter.
// Memory-bound: ~1 GiB total HBM traffic -> ~46 us floor at 23.3 TB/s.
// ---------------------------------------------------------------------------
__global__ __launch_bounds__(256) void add_relpos(
    const float* __restrict__ attn, const float* __restrict__ rel_h,
    const float* __restrict__ rel_w, float* __restrict__ out)
{
  __shared__ float sh[64];
  __shared__ float sw[64];

  const int  p     = blockIdx.x;            // b*4096 + h*64 + w
  const long base  = (long)p * 4096;        // 64*64 slice
  const long rbase = (long)p * 64;
  const int  tid   = threadIdx.x;

  if (tid < 64)        sh[tid]      = rel_h[rbase + tid];
  else if (tid < 128)  sw[tid - 64] = rel_w[rbase + (tid - 64)];
  __syncthreads();

  const float4* __restrict__ in4  = (const float4*)(attn + base);
  float4* __restrict__       out4 = (float4*)(out + base);

  #pragma unroll
  for (int it = 0; it < 4; ++it) {
    const int i   = it * 256 + tid;         // 0..1023
    const int kh  = i >> 4;
    const int kw4 = (i & 15) << 2;
    float4 v = in4[i];
    const float rh = sh[kh];
    v.x += rh + sw[kw4 + 0];
    v.y += rh + sw[kw4 + 1];
    v.z += rh + sw[kw4 + 2];
    v.w += rh + sw[kw4 + 3];
    out4[i] = v;
  }
}

// ---------------------------------------------------------------------------
// Fallback (no workspace): compute rel rows on the fly per slice, then add.
// ---------------------------------------------------------------------------
__global__ __launch_bounds__(256) void fused_fallback(
    const float* __restrict__ attn, const float* __restrict__ q,
    const float* __restrict__ rph, const float* __restrict__ rpw,
    float* __restrict__ out)
{
  __shared__ float qs[64];
  __shared__ float sh[64];
  __shared__ float sw[64];

  const int  p    = blockIdx.x;             // b*4096 + h*64 + w
  const int  h    = (p >> 6) & 63;
  const int  w    = p & 63;
  const long base = (long)p * 4096;
  const int  tid  = threadIdx.x;

  if (tid < 64) qs[tid] = q[(long)p * 64 + tid];
  __syncthreads();

  if (tid < 64) {
    const float* __restrict__ r = rph + (long)(h - tid + 63) * 64;
    float acc = 0.f;
    #pragma unroll 8
    for (int c = 0; c < 64; ++c) acc += qs[c] * r[c];
    sh[tid] = acc;
  } else if (tid < 128) {
    const int k = tid - 64;
    const float* __restrict__ r = rpw + (long)(w - k + 63) * 64;
    float acc = 0.f;
    #pragma unroll 8
    for (int c = 0; c < 64; ++c) acc += qs[c] * r[c];
    sw[k] = acc;
  }
  __syncthreads();

  const float4* __restrict__ in4  = (const float4*)(attn + base);
  float4* __restrict__       out4 = (float4*)(out + base);
  #pragma unroll
  for (int it = 0; it < 4; ++it) {
    const int i   = it * 256 + tid;
    const int kh  = i >> 4;
    const int kw4 = (i & 15) << 2;
    float4 v = in4[i];
    const float rh = sh[kh];
    v.x += rh + sw[kw4 + 0];
    v.y += rh + sw[kw4 + 1];
    v.z += rh + sw[kw4 + 2];
    v.w += rh + sw[kw4 + 3];
    out4[i] = v;
  }
}

extern "C" void kernel_launch(void* const* d_in, const int* in_sizes, int n_in,
                              void* d_out, int out_size, void* d_ws, size_t ws_size,
                              hipStream_t stream) {
  const float* attn = (const float*)d_in[0];   // (B, 4096, 4096)
  const float* q    = (const float*)d_in[1];   // (B, 4096, 64)
  const float* rph  = (const float*)d_in[2];   // (127, 64)
  const float* rpw  = (const float*)d_in[3];   // (127, 64)
  float* out = (float*)d_out;

  const int B = in_sizes[1] / (4096 * 64);     // derive batch from queries
  const size_t rel_elems = (size_t)B * 4096 * 64;
  const size_t ws_need   = 2 * rel_elems * sizeof(float);   // 16 MiB for B=8

  if (d_ws != nullptr && ws_size >= ws_need) {
    float* rel_h = (float*)d_ws;
    float* rel_w = rel_h + rel_elems;
    // rel_h: f=h (f_stride=4096), M=w (m_stride=64)
    relpos_gemm_wmma<<<B * 64, 256, 0, stream>>>(q, rph, rel_h, 4096, 64);
    // rel_w: f=w (f_stride=64), M=h (m_stride=4096)
    relpos_gemm_wmma<<<B * 64, 256, 0, stream>>>(q, rpw, rel_w, 64, 4096);
    add_relpos<<<B * 4096, 256, 0, stream>>>(attn, rel_h, rel_w, out);
  } else {
    fused_fallback<<<B * 4096, 256, 0, stream>>>(attn, q, rph, rpw, out);
  }
}